// QuaternionSSM_18588618457727
// MI455X (gfx1250) — compile-verified
//
#include <hip/hip_runtime.h>
#include <math.h>

#define BB 2
#define LL 1024
#define DMQ 128
#define D_STATE 64
#define D_CONV 4
#define D_INNER 256
#define D4 1024
#define DT_RANK 8
#define MROWS (BB*LL)                    // 2048
#define XPROJ_N (DT_RANK + 2*D_STATE)    // 136

typedef __attribute__((ext_vector_type(2))) float v2f;
typedef __attribute__((ext_vector_type(8))) float v8f;

// Hamilton-product block structure: Wbig[row block p (input comp)][col block q (output comp)]
__constant__ int   kQIdx[4][4] = {{0,1,2,3},{1,0,3,2},{2,3,0,1},{3,2,1,0}};
__constant__ float kQSgn[4][4] = {{ 1.f, 1.f, 1.f, 1.f},
                                  {-1.f, 1.f,-1.f, 1.f},
                                  {-1.f, 1.f, 1.f,-1.f},
                                  {-1.f,-1.f, 1.f, 1.f}};

// ---------- pack q_{r,i,j,k} [M,128] each -> Qpack [M, 512] ----------
__global__ void pack_q_kernel(const float* __restrict__ qr, const float* __restrict__ qi,
                              const float* __restrict__ qj, const float* __restrict__ qk,
                              float* __restrict__ Qp) {
  int idx = blockIdx.x * blockDim.x + threadIdx.x;
  if (idx >= MROWS * 4 * DMQ) return;
  int c = idx % (4 * DMQ);
  int m = idx / (4 * DMQ);
  int p = c / DMQ, k = c % DMQ;
  const float* src = (p == 0) ? qr : (p == 1) ? qi : (p == 2) ? qj : qk;
  Qp[idx] = src[m * DMQ + k];
}

// ---------- pack 4 quaternion weights [KB,NB] -> TRANSPOSED Wbig^T [4*NB, 4*KB] ----------
__global__ void pack_wT_kernel(const float* __restrict__ w0, const float* __restrict__ w1,
                               const float* __restrict__ w2, const float* __restrict__ w3,
                               float* __restrict__ WbT, int KB, int NB) {
  int idx = blockIdx.x * blockDim.x + threadIdx.x;   // index over [4*NB, 4*KB] output
  int krows = 4 * KB;
  if (idx >= 4 * NB * krows) return;
  int row = idx / krows;                             // output row = big col index (n-dim)
  int col = idx % krows;                             // output col = big row index (k-dim)
  int q = row / NB, n = row % NB;
  int p = col / KB, k = col % KB;
  int wi = kQIdx[p][q];
  const float* w = (wi == 0) ? w0 : (wi == 1) ? w1 : (wi == 2) ? w2 : w3;
  WbT[idx] = kQSgn[p][q] * w[k * NB + n];
}

// ---------- generic transpose: B[K,N] -> BT[N,K] ----------
__global__ void transpose_kernel(const float* __restrict__ B, float* __restrict__ BT,
                                 int K, int N) {
  int idx = blockIdx.x * blockDim.x + threadIdx.x;
  if (idx >= K * N) return;
  int k = idx / N, n = idx % N;
  BT[(size_t)n * K + k] = B[idx];
}

// ---------- f32 WMMA GEMM: C[M,N] = A[M,K] @ B[K,N] with B given TRANSPOSED (BT[N,K]).
// 32x32 (2x2 tiles) per wave. B row index is CLAMPED (not predicated) so the inner loop
// runs with full EXEC and no divergence; junk columns are never stored. Both A and B
// fragment pairs are K-contiguous -> b64 loads.
__global__ void gemm_wmma_f32(const float* __restrict__ A, int lda,
                              const float* __restrict__ BT,
                              float* __restrict__ C, int ldc,
                              int M, int N, int K, int ntn, int nwaves) {
  int wave = blockIdx.x * (blockDim.x >> 5) + (threadIdx.x >> 5);
  if (wave >= nwaves) return;                 // wave-uniform exit: EXEC all-1 for WMMA
  int lane = threadIdx.x & 31;
  int tm = wave / ntn, tn = wave % ntn;
  int m0 = tm * 32, n0 = tn * 32;
  int half = lane >> 4;                       // 0: K+0/K+1 ; 1: K+2/K+3
  int l16  = lane & 15;

  const float* arow0 = A + (size_t)(m0 + l16) * lda;        // rows m0..m0+15
  const float* arow1 = arow0 + (size_t)16 * lda;            // rows m0+16..m0+31
  int nb0 = n0 + l16;
  int nb1 = nb0 + 16;
  int nc0 = (nb0 < N) ? nb0 : (N - 1);                      // clamped load columns
  int nc1 = (nb1 < N) ? nb1 : (N - 1);
  const float* brow0 = BT + (size_t)nc0 * K;
  const float* brow1 = BT + (size_t)nc1 * K;

  v8f acc00 = {}, acc01 = {}, acc10 = {}, acc11 = {};
#pragma unroll 4
  for (int k = 0; k < K; k += 4) {
    int ka = k + 2 * half;                                  // even -> 8B-aligned pairs
    v2f a0, a1, b0, b1;
    a0.x = arow0[ka];  a0.y = arow0[ka + 1];
    a1.x = arow1[ka];  a1.y = arow1[ka + 1];
    b0.x = brow0[ka];  b0.y = brow0[ka + 1];
    b1.x = brow1[ka];  b1.y = brow1[ka + 1];
    acc00 = __builtin_amdgcn_wmma_f32_16x16x4_f32(false, a0, false, b0, (short)0, acc00, false, false);
    acc01 = __builtin_amdgcn_wmma_f32_16x16x4_f32(false, a0, false, b1, (short)0, acc01, false, false);
    acc10 = __builtin_amdgcn_wmma_f32_16x16x4_f32(false, a1, false, b0, (short)0, acc10, false, false);
    acc11 = __builtin_amdgcn_wmma_f32_16x16x4_f32(false, a1, false, b1, (short)0, acc11, false, false);
  }

  // C/D layout: VGPR r -> row r (lanes 0-15), row r+8 (lanes 16-31)
  int mr0 = m0 + (half ? 8 : 0);
  int mr1 = mr0 + 16;
  if (nb0 < N) {
#pragma unroll
    for (int r = 0; r < 8; ++r) C[(size_t)(mr0 + r) * ldc + nb0] = acc00[r];
#pragma unroll
    for (int r = 0; r < 8; ++r) C[(size_t)(mr1 + r) * ldc + nb0] = acc10[r];
  }
  if (nb1 < N) {
#pragma unroll
    for (int r = 0; r < 8; ++r) C[(size_t)(mr0 + r) * ldc + nb1] = acc01[r];
#pragma unroll
    for (int r = 0; r < 8; ++r) C[(size_t)(mr1 + r) * ldc + nb1] = acc11[r];
  }
}

// ---------- causal depthwise conv1d (k=4) + bias + silu; x = xz[:, :D4] ----------
__global__ void conv_silu_kernel(const float* __restrict__ xz, const float* __restrict__ conv_w,
                                 const float* __restrict__ conv_b, float* __restrict__ xconv) {
  int idx = blockIdx.x * blockDim.x + threadIdx.x;
  if (idx >= MROWS * D4) return;
  int d = idx & (D4 - 1);
  int m = idx >> 10;                          // b*L + l
  int l = m & (LL - 1);
  int b = m >> 10;
  float acc = conv_b[d];
#pragma unroll
  for (int t = 0; t < D_CONV; ++t) {
    int ls = l - (D_CONV - 1) + t;
    if (ls >= 0)
      acc += conv_w[d * D_CONV + t] * xz[(size_t)((b << 10) + ls) * (2 * D4) + d];
  }
  xconv[idx] = acc / (1.f + __expf(-acc));    // silu
}

// ---------- selective scan: one wave per (b,d); lanes hold states n and n+32 ----------
__global__ void scan_kernel(const float* __restrict__ xz, const float* __restrict__ xconv,
                            const float* __restrict__ xdbl, const float* __restrict__ dpre,
                            const float* __restrict__ dtb, const float* __restrict__ A_log,
                            const float* __restrict__ D_skip, float* __restrict__ y) {
  int wid  = blockIdx.x * (blockDim.x >> 5) + (threadIdx.x >> 5);
  if (wid >= BB * D4) return;
  int lane = threadIdx.x & 31;
  int b = wid >> 10;
  int d = wid & (D4 - 1);
  float a0 = -__expf(A_log[d * D_STATE + lane]);
  float a1 = -__expf(A_log[d * D_STATE + lane + 32]);
  float bias = dtb[d];
  float dsk  = D_skip[d];
  float h0 = 0.f, h1 = 0.f;
  for (int t = 0; t < LL; ++t) {
    int m = b * LL + t;
    float dp = dpre[(size_t)m * D4 + d] + bias;
    float delta = (dp > 20.f) ? dp : log1pf(__expf(dp));   // softplus
    float xv = xconv[(size_t)m * D4 + d];
    const float* row = xdbl + (size_t)m * XPROJ_N;
    float Bn0 = row[DT_RANK + lane];
    float Bn1 = row[DT_RANK + lane + 32];
    float Cn0 = row[DT_RANK + D_STATE + lane];
    float Cn1 = row[DT_RANK + D_STATE + lane + 32];
    float dbx = delta * xv;
    h0 = __expf(delta * a0) * h0 + dbx * Bn0;
    h1 = __expf(delta * a1) * h1 + dbx * Bn1;
    float s = h0 * Cn0 + h1 * Cn1;
#pragma unroll
    for (int off = 16; off; off >>= 1) s += __shfl_xor(s, off, 32);
    if (lane == 0) {
      float zv = xz[(size_t)m * (2 * D4) + D4 + d];
      float sz = zv / (1.f + __expf(-zv));                 // silu(z)
      y[(size_t)m * D4 + d] = (s + xv * dsk) * sz;
    }
  }
}

extern "C" void kernel_launch(void* const* d_in, const int* in_sizes, int n_in,
                              void* d_out, int out_size, void* d_ws, size_t ws_size,
                              hipStream_t stream) {
  const float* qr      = (const float*)d_in[0];
  const float* qi      = (const float*)d_in[1];
  const float* qj      = (const float*)d_in[2];
  const float* qk      = (const float*)d_in[3];
  const float* inWr    = (const float*)d_in[4];
  const float* inWi    = (const float*)d_in[5];
  const float* inWj    = (const float*)d_in[6];
  const float* inWk    = (const float*)d_in[7];
  const float* conv_w  = (const float*)d_in[8];
  const float* conv_b  = (const float*)d_in[9];
  const float* xprojW  = (const float*)d_in[10];
  const float* dtprojW = (const float*)d_in[11];
  const float* dtprojb = (const float*)d_in[12];
  const float* A_log   = (const float*)d_in[13];
  const float* D_skip  = (const float*)d_in[14];
  const float* outWr   = (const float*)d_in[15];
  const float* outWi   = (const float*)d_in[16];
  const float* outWj   = (const float*)d_in[17];
  const float* outWk   = (const float*)d_in[18];
  float* out = (float*)d_out;

  // workspace layout (floats)
  float* ws        = (float*)d_ws;
  float* WbigT_in  = ws;                                    // [2048, 512]
  float* Qp        = WbigT_in  + (size_t)2048 * 512;        // [2048, 512]
  float* xz        = Qp        + (size_t)2048 * 512;        // [2048, 2048]
  float* xconv     = xz        + (size_t)2048 * 2048;       // [2048, 1024]
  float* xdbl      = xconv     + (size_t)2048 * 1024;       // [2048, 136]
  float* dpre      = xdbl      + (size_t)2048 * XPROJ_N;    // [2048, 1024]
  float* WbigT_out = dpre      + (size_t)2048 * 1024;       // [512, 1024]
  float* ybig      = WbigT_out + (size_t)512 * 1024;        // [2048, 1024]
  float* xprojWT   = ybig      + (size_t)2048 * 1024;       // [136, 1024]
  float* dtprojWT  = xprojWT   + (size_t)XPROJ_N * 1024;    // [1024, 8]

  // 1) packing (weights packed directly in transposed [N,K] form)
  {
    int n = MROWS * 4 * DMQ;
    pack_q_kernel<<<(n + 255) / 256, 256, 0, stream>>>(qr, qi, qj, qk, Qp);
  }
  {
    int n = 2048 * 512;
    pack_wT_kernel<<<(n + 255) / 256, 256, 0, stream>>>(inWr, inWi, inWj, inWk,
                                                        WbigT_in, DMQ, 2 * D_INNER);
  }
  {
    int n = 512 * 1024;
    pack_wT_kernel<<<(n + 255) / 256, 256, 0, stream>>>(outWr, outWi, outWj, outWk,
                                                        WbigT_out, D_INNER, DMQ);
  }
  {
    int n = D4 * XPROJ_N;
    transpose_kernel<<<(n + 255) / 256, 256, 0, stream>>>(xprojW, xprojWT, D4, XPROJ_N);
  }
  {
    int n = DT_RANK * D4;
    transpose_kernel<<<(n + 255) / 256, 256, 0, stream>>>(dtprojW, dtprojWT, DT_RANK, D4);
  }

  auto launch_gemm = [&](const float* A, int lda, const float* BT,
                         float* C, int ldc, int M, int N, int K) {
    int ntn = (N + 31) / 32;                  // 32-wide column tiles
    int nwaves = (M / 32) * ntn;
    int wpb = 4;                              // 4 waves / block (128 threads)
    int blocks = (nwaves + wpb - 1) / wpb;
    gemm_wmma_f32<<<blocks, wpb * 32, 0, stream>>>(A, lda, BT, C, ldc,
                                                   M, N, K, ntn, nwaves);
  };

  // 2) in_proj as one Hamilton-packed GEMM: xz = Qp[2048,512] @ Wbig_in[512,2048]
  launch_gemm(Qp, 512, WbigT_in, xz, 2048, MROWS, 2048, 512);

  // 3) depthwise causal conv + silu
  {
    int n = MROWS * D4;
    conv_silu_kernel<<<(n + 255) / 256, 256, 0, stream>>>(xz, conv_w, conv_b, xconv);
  }

  // 4) x_dbl = xconv[2048,1024] @ xproj_W[1024,136]
  launch_gemm(xconv, D4, xprojWT, xdbl, XPROJ_N, MROWS, XPROJ_N, D4);

  // 5) delta_pre = x_dbl[:, :8] @ dtproj_W[8,1024]  (bias+softplus fused into scan)
  launch_gemm(xdbl, XPROJ_N, dtprojWT, dpre, D4, MROWS, D4, DT_RANK);

  // 6) selective scan, fused epilogue (s + x*D)*silu(z) -> ybig
  {
    int waves = BB * D4;                      // 2048 waves
    int wpb = 8;                              // 256 threads / block
    scan_kernel<<<(waves + wpb - 1) / wpb, wpb * 32, 0, stream>>>(
        xz, xconv, xdbl, dpre, dtprojb, A_log, D_skip, ybig);
  }

  // 7) out_proj as one Hamilton-packed GEMM straight into d_out
  launch_gemm(ybig, D4, WbigT_out, out, 4 * DMQ, MROWS, 4 * DMQ, D4);
}